// DeformableConditionalPositionalEncoding2D_22213570855632
// MI455X (gfx1250) — compile-verified
//
#include <hip/hip_runtime.h>
#include <hip/hip_bf16.h>

typedef __attribute__((ext_vector_type(2))) float v2f;
typedef __attribute__((ext_vector_type(8))) float v8f;

#define Bn   4
#define Cc   256
#define Hh   48
#define Ww   160
#define HWn  7680        // Hh*Ww
#define HIDn 256
#define NIMG (Cc * HWn)          // per-batch x elements
#define NTOT (Bn * Cc * HWn)     // 7864320
#define OFF_ELEMS (Bn * 18 * HWn)   // 552960
#define WT_ELEMS  (9 * 256 * 256)   // 589824
#define WP_ELEMS  (256 * 256)

// ---------------------------------------------------------------------------
// Kernel 0: weight transposes for coalesced GEMM staging.
//   Wt[kk][c][o] = deform_w[o][c][kk]   (for deform GEMM, A=weights read by o)
//   Wp[c][o]     = proj_w[o][c]
// ---------------------------------------------------------------------------
__global__ __launch_bounds__(256) void prep_weights_kernel(
    const float* __restrict__ dw, const float* __restrict__ pw,
    float* __restrict__ Wt, float* __restrict__ Wp) {
  int i = blockIdx.x * 256 + threadIdx.x;
  if (i < WT_ELEMS) {
    int o  = i & 255;
    int c  = (i >> 8) & 255;
    int kk = i >> 16;
    Wt[i] = dw[(o * 256 + c) * 9 + kk];
  } else {
    int j = i - WT_ELEMS;
    if (j < WP_ELEMS) {
      int o = j & 255;
      int c = j >> 8;
      Wp[c * 256 + o] = pw[o * 256 + c];
    }
  }
}

// ---------------------------------------------------------------------------
// Kernel 1: offset predictor conv 3x3, C=256 -> 18, pad 1.  N=18 is too skinny
// for WMMA; plain VALU, coalesced over w.
// ---------------------------------------------------------------------------
__global__ __launch_bounds__(160) void off_conv_kernel(
    const float* __restrict__ x, const float* __restrict__ w,
    const float* __restrict__ bias, float* __restrict__ off) {
  const int ww = threadIdx.x;
  const int hh = blockIdx.x;
  const int oc = blockIdx.y;
  const int b  = blockIdx.z;
  float acc = bias[oc];
  const float* xb = x + (size_t)b * NIMG;
  const float* wb = w + (size_t)oc * Cc * 9;
  for (int c = 0; c < Cc; ++c) {
    const float* xc = xb + c * HWn;
    const float* wc = wb + c * 9;
#pragma unroll
    for (int ky = 0; ky < 3; ++ky) {
      int y = hh + ky - 1;
      if ((unsigned)y >= (unsigned)Hh) continue;
#pragma unroll
      for (int kx = 0; kx < 3; ++kx) {
        int xx = ww + kx - 1;
        if ((unsigned)xx >= (unsigned)Ww) continue;
        acc = fmaf(xc[y * Ww + xx], wc[ky * 3 + kx], acc);
      }
    }
  }
  off[((size_t)(b * 18 + oc)) * HWn + hh * Ww + ww] = acc;
}

// ---------------------------------------------------------------------------
// Kernel 2: deformable conv as fused bilinear-im2col + WMMA f32 GEMM.
//   D[o=256][pix=64 tile], K = 9 taps * 256 channels, WMMA 16x16x4 f32.
//   8 waves: wave wv owns o in [32*wv, 32*wv+32), all 4 pixel fragments.
// Fragment layouts per ISA 7.12.2:
//   A 16x4 f32 : lane l holds M=l%16 ; VGPR0 = K=(2*hi), VGPR1 = K=(2*hi+1)
//   B 4x16     : VGPR0 lanes<16 -> K=0 N=lane ; lanes>=16 -> K=2 ; VGPR1 K=1/3
//   C/D 16x16  : VGPR r -> M = r + 8*hi, N = lane%16
// ---------------------------------------------------------------------------
__global__ __launch_bounds__(256) void deform_gemm_kernel(
    const float* __restrict__ x, const float* __restrict__ off,
    const float* __restrict__ Wt, float* __restrict__ h_out) {
  __shared__ float W_s[32][264];   // weights  [k within chunk][o], padded
  __shared__ float S_s[32][72];    // samples  [k within chunk][pix], padded
  __shared__ int   sy0[9][64];
  __shared__ int   sx0[9][64];
  __shared__ float sfy[9][64];
  __shared__ float sfx[9][64];

  const int b        = blockIdx.y;
  const int tileBase = blockIdx.x * 64;
  const int t    = threadIdx.x;
  const int lane = t & 31;
  const int wv   = t >> 5;
  const int l16  = lane & 15;
  const int hi   = lane >> 4;
  const int o0w  = wv * 32;
  const int pixl = t & 63;     // staging role: pixel
  const int cg   = t >> 6;     // staging role: channel sub-group 0..3

  // ---- precompute 9 bilinear taps per pixel ----
  for (int i = t; i < 9 * 64; i += 256) {
    int kk  = i >> 6;
    int pix = i & 63;
    int p   = tileBase + pix;
    int hh  = p / Ww;
    int ww  = p - hh * Ww;
    float dy = off[((size_t)(b * 18 + kk * 2))     * HWn + p];
    float dx = off[((size_t)(b * 18 + kk * 2 + 1)) * HWn + p];
    float py = (float)(hh - 1 + kk / 3) + dy;
    float px = (float)(ww - 1 + kk % 3) + dx;
    float y0f = floorf(py);
    float x0f = floorf(px);
    sy0[kk][pix] = (int)y0f;
    sx0[kk][pix] = (int)x0f;
    sfy[kk][pix] = py - y0f;
    sfx[kk][pix] = px - x0f;
  }
  __syncthreads();

  v8f acc[2][4];
  const v8f vz = {0.f, 0.f, 0.f, 0.f, 0.f, 0.f, 0.f, 0.f};
#pragma unroll
  for (int i = 0; i < 2; ++i)
#pragma unroll
    for (int j = 0; j < 4; ++j) acc[i][j] = vz;

  for (int kk = 0; kk < 9; ++kk) {
    // per-thread tap data for its staging pixel
    const int   y0 = sy0[kk][pixl];
    const int   x0 = sx0[kk][pixl];
    const float fy = sfy[kk][pixl];
    const float fx = sfx[kk][pixl];
    const float w00 = (1.f - fy) * (1.f - fx);
    const float w01 = (1.f - fy) * fx;
    const float w10 = fy * (1.f - fx);
    const float w11 = fy * fx;
    const bool vy0 = (unsigned)y0       < (unsigned)Hh;
    const bool vy1 = (unsigned)(y0 + 1) < (unsigned)Hh;
    const bool vx0 = (unsigned)x0       < (unsigned)Ww;
    const bool vx1 = (unsigned)(x0 + 1) < (unsigned)Ww;

    for (int cb = 0; cb < 8; ++cb) {
      const int c0 = cb * 32;
      // stage weights (coalesced over o)
      const float* wsrc = Wt + ((size_t)(kk * 256 + c0)) * 256;
#pragma unroll
      for (int i = 0; i < 32; ++i) W_s[i][t] = wsrc[i * 256 + t];
      // stage bilinear samples
#pragma unroll
      for (int j = 0; j < 8; ++j) {
        int ci = cg + 4 * j;
        const float* xb = x + ((size_t)(b * Cc + c0 + ci)) * HWn;
        float v00 = (vy0 && vx0) ? xb[y0 * Ww + x0]           : 0.f;
        float v01 = (vy0 && vx1) ? xb[y0 * Ww + x0 + 1]       : 0.f;
        float v10 = (vy1 && vx0) ? xb[(y0 + 1) * Ww + x0]     : 0.f;
        float v11 = (vy1 && vx1) ? xb[(y0 + 1) * Ww + x0 + 1] : 0.f;
        S_s[ci][pixl] = v00 * w00 + v01 * w01 + v10 * w10 + v11 * w11;
      }
      __syncthreads();
      // WMMA over this 32-deep K chunk
#pragma unroll
      for (int kc = 0; kc < 8; ++kc) {
        const int kb = kc * 4 + 2 * hi;
        v2f a0, a1;
        a0.x = W_s[kb][o0w + l16];        a0.y = W_s[kb + 1][o0w + l16];
        a1.x = W_s[kb][o0w + 16 + l16];   a1.y = W_s[kb + 1][o0w + 16 + l16];
#pragma unroll
        for (int pf = 0; pf < 4; ++pf) {
          v2f bf;
          bf.x = S_s[kb][pf * 16 + l16];
          bf.y = S_s[kb + 1][pf * 16 + l16];
          acc[0][pf] = __builtin_amdgcn_wmma_f32_16x16x4_f32(
              false, a0, false, bf, (short)0, acc[0][pf], false, false);
          acc[1][pf] = __builtin_amdgcn_wmma_f32_16x16x4_f32(
              false, a1, false, bf, (short)0, acc[1][pf], false, false);
        }
      }
      __syncthreads();
    }
  }

  // epilogue: D element (M = r + 8*hi, N = l16) -> h_out[b][o][pix]
#pragma unroll
  for (int of = 0; of < 2; ++of)
#pragma unroll
    for (int pf = 0; pf < 4; ++pf)
#pragma unroll
      for (int r = 0; r < 8; ++r) {
        int o = o0w + of * 16 + r + hi * 8;
        int p = tileBase + pf * 16 + l16;
        h_out[((size_t)(b * HIDn + o)) * HWn + p] = acc[of][pf][r];
      }
}

// ---------------------------------------------------------------------------
// Kernel 3: GroupNorm statistics (population var), one block per (b, group).
// ---------------------------------------------------------------------------
__global__ __launch_bounds__(256) void gn_stats_kernel(
    const float* __restrict__ h, float* __restrict__ stats) {
  const int bg = blockIdx.x;
  const int b = bg >> 4, g = bg & 15;
  const float* base = h + ((size_t)(b * HIDn + g * 16)) * HWn;
  float s = 0.f, ss = 0.f;
  for (int i = threadIdx.x; i < 16 * HWn; i += 256) {
    float v = base[i];
    s += v;
    ss += v * v;
  }
  __shared__ float rs[256], rss[256];
  rs[threadIdx.x] = s;
  rss[threadIdx.x] = ss;
  __syncthreads();
  for (int st = 128; st > 0; st >>= 1) {
    if (threadIdx.x < st) {
      rs[threadIdx.x]  += rs[threadIdx.x + st];
      rss[threadIdx.x] += rss[threadIdx.x + st];
    }
    __syncthreads();
  }
  if (threadIdx.x == 0) {
    const float inv = 1.f / (16.f * HWn);
    float mu  = rs[0] * inv;
    float var = rss[0] * inv - mu * mu;
    stats[bg * 2]     = mu;
    stats[bg * 2 + 1] = rsqrtf(var + 1e-5f);
  }
}

// ---------------------------------------------------------------------------
// Kernel 4: fused GN-normalize + SiLU + 1x1 proj (WMMA GEMM) + bias +
// residual.  Same tile structure as kernel 2, K = 256.
// ---------------------------------------------------------------------------
__global__ __launch_bounds__(256) void proj_gemm_kernel(
    const float* __restrict__ x, const float* __restrict__ h,
    const float* __restrict__ Wp, const float* __restrict__ gn_w,
    const float* __restrict__ gn_b, const float* __restrict__ stats,
    const float* __restrict__ proj_b, float* __restrict__ out) {
  __shared__ float W_s[32][264];
  __shared__ float S_s[32][72];

  const int b        = blockIdx.y;
  const int tileBase = blockIdx.x * 64;
  const int t    = threadIdx.x;
  const int lane = t & 31;
  const int wv   = t >> 5;
  const int l16  = lane & 15;
  const int hi   = lane >> 4;
  const int o0w  = wv * 32;
  const int pixl = t & 63;
  const int cg   = t >> 6;

  v8f acc[2][4];
  const v8f vz = {0.f, 0.f, 0.f, 0.f, 0.f, 0.f, 0.f, 0.f};
#pragma unroll
  for (int i = 0; i < 2; ++i)
#pragma unroll
    for (int j = 0; j < 4; ++j) acc[i][j] = vz;

  for (int cb = 0; cb < 8; ++cb) {
    const int c0 = cb * 32;
#pragma unroll
    for (int i = 0; i < 32; ++i) W_s[i][t] = Wp[(c0 + i) * 256 + t];
#pragma unroll
    for (int j = 0; j < 8; ++j) {
      int ci = cg + 4 * j;
      int c  = c0 + ci;
      int g  = c >> 4;
      float mu = stats[(b * 16 + g) * 2];
      float rs = stats[(b * 16 + g) * 2 + 1];
      float hv = h[((size_t)(b * HIDn + c)) * HWn + tileBase + pixl];
      float hn = (hv - mu) * rs * gn_w[c] + gn_b[c];
      S_s[ci][pixl] = hn / (1.f + __expf(-hn));   // SiLU
    }
    __syncthreads();
#pragma unroll
    for (int kc = 0; kc < 8; ++kc) {
      const int kb = kc * 4 + 2 * hi;
      v2f a0, a1;
      a0.x = W_s[kb][o0w + l16];        a0.y = W_s[kb + 1][o0w + l16];
      a1.x = W_s[kb][o0w + 16 + l16];   a1.y = W_s[kb + 1][o0w + 16 + l16];
#pragma unroll
      for (int pf = 0; pf < 4; ++pf) {
        v2f bf;
        bf.x = S_s[kb][pf * 16 + l16];
        bf.y = S_s[kb + 1][pf * 16 + l16];
        acc[0][pf] = __builtin_amdgcn_wmma_f32_16x16x4_f32(
            false, a0, false, bf, (short)0, acc[0][pf], false, false);
        acc[1][pf] = __builtin_amdgcn_wmma_f32_16x16x4_f32(
            false, a1, false, bf, (short)0, acc[1][pf], false, false);
      }
    }
    __syncthreads();
  }

  // epilogue: pe = acc + proj_b ; out0 = x + pe ; out1 = pe
#pragma unroll
  for (int of = 0; of < 2; ++of)
#pragma unroll
    for (int pf = 0; pf < 4; ++pf)
#pragma unroll
      for (int r = 0; r < 8; ++r) {
        int o = o0w + of * 16 + r + hi * 8;
        int p = tileBase + pf * 16 + l16;
        size_t idx = ((size_t)(b * Cc + o)) * HWn + p;
        float pe = acc[of][pf][r] + proj_b[o];
        out[idx]        = x[idx] + pe;
        out[NTOT + idx] = pe;
      }
}

// ---------------------------------------------------------------------------
extern "C" void kernel_launch(void* const* d_in, const int* in_sizes, int n_in,
                              void* d_out, int out_size, void* d_ws,
                              size_t ws_size, hipStream_t stream) {
  (void)in_sizes; (void)n_in; (void)out_size; (void)ws_size;
  const float* x        = (const float*)d_in[0];
  const float* off_w    = (const float*)d_in[1];
  const float* off_b    = (const float*)d_in[2];
  const float* deform_w = (const float*)d_in[3];
  const float* gn_w     = (const float*)d_in[4];
  const float* gn_b     = (const float*)d_in[5];
  const float* proj_w   = (const float*)d_in[6];
  const float* proj_b   = (const float*)d_in[7];
  float* out = (float*)d_out;
  float* ws  = (float*)d_ws;

  float* off_out = ws;                       // 552960
  float* h_out   = off_out + OFF_ELEMS;      // 7864320
  float* Wt      = h_out + NTOT;             // 589824
  float* Wp      = Wt + WT_ELEMS;            // 65536
  float* stats   = Wp + WP_ELEMS;            // 128

  prep_weights_kernel<<<(WT_ELEMS + WP_ELEMS) / 256, 256, 0, stream>>>(
      deform_w, proj_w, Wt, Wp);
  off_conv_kernel<<<dim3(Hh, 18, Bn), 160, 0, stream>>>(x, off_w, off_b,
                                                        off_out);
  deform_gemm_kernel<<<dim3(HWn / 64, Bn), 256, 0, stream>>>(x, off_out, Wt,
                                                             h_out);
  gn_stats_kernel<<<Bn * 16, 256, 0, stream>>>(h_out, stats);
  proj_gemm_kernel<<<dim3(HWn / 64, Bn), 256, 0, stream>>>(
      x, h_out, Wp, gn_w, gn_b, stats, proj_b, out);
}